// GraphPolicy_1881195675940
// MI455X (gfx1250) — compile-verified
//
#include <hip/hip_runtime.h>

typedef float v2f __attribute__((ext_vector_type(2)));
typedef float v8f __attribute__((ext_vector_type(8)));

#define N_NODES 100000
#define N_EDGES 1600000
#define E_TOT   (N_EDGES + N_NODES)
#define IN_DIM  128
#define HID_DIM 64
#define OUT_DIM 32
#define NEG_SLOPE 0.2f

// ---------- monotone float <-> uint encoding for atomicMax on floats ----------
__device__ __forceinline__ unsigned enc_f32(float f) {
    unsigned u = __float_as_uint(f);
    return (u & 0x80000000u) ? ~u : (u | 0x80000000u);
}
__device__ __forceinline__ float dec_f32(unsigned u) {
    u = (u & 0x80000000u) ? (u & 0x7fffffffu) : ~u;
    return __uint_as_float(u);
}

// ---------- WMMA f32 GEMM: H[M,ND] = X[M,KD] @ W[KD,ND] ----------
// One wave per 16x16 output tile. KD % 4 == 0, ND % 16 == 0, M % 16 == 0.
template <int KD, int ND>
__global__ void gemm_wmma_f32(const float* __restrict__ X,
                              const float* __restrict__ W,
                              float* __restrict__ H, int M) {
    const int wave  = (blockIdx.x * blockDim.x + threadIdx.x) >> 5;
    const int lane  = threadIdx.x & 31;
    const int ntile = ND / 16;
    const int mt = wave / ntile;
    const int nt = wave - mt * ntile;
    if (mt * 16 >= M) return;              // wave-uniform: EXEC stays all-ones

    const int m0   = mt * 16;
    const int n0   = nt * 16;
    const int half = lane >> 4;            // 0: lanes 0-15, 1: lanes 16-31
    const int lr   = lane & 15;

    v8f acc = {};
    const float* xrow = X + (size_t)(m0 + lr) * KD;
    for (int k0 = 0; k0 < KD; k0 += 4) {
        const int ka = k0 + 2 * half;      // this half-wave covers K = ka, ka+1
        v2f a = *(const v2f*)(xrow + ka);  // A[16x4]: {K even, K odd} per lane
        v2f b;
        b.x = W[(size_t)(ka + 0) * ND + n0 + lr];  // B[4x16] mirrors A's K split
        b.y = W[(size_t)(ka + 1) * ND + n0 + lr];
        acc = __builtin_amdgcn_wmma_f32_16x16x4_f32(
            /*neg_a=*/false, a, /*neg_b=*/false, b,
            /*c_mod=*/(short)0, acc, /*reuse_a=*/false, /*reuse_b=*/false);
    }
    // C/D layout: N = lane&15, M = vgpr + 8*(lane>>4)
    float* hrow = H + (size_t)m0 * ND + n0 + lr;
#pragma unroll
    for (int r = 0; r < 8; ++r)
        hrow[(size_t)(r + 8 * half) * ND] = acc[r];
}

// ---------- per-node attention scores: alpha = h . a ----------
template <int F>
__global__ void alpha_kernel(const float* __restrict__ H,
                             const float* __restrict__ a_src,
                             const float* __restrict__ a_dst,
                             float* __restrict__ As, float* __restrict__ Ad, int N) {
    int n = blockIdx.x * blockDim.x + threadIdx.x;
    if (n >= N) return;
    const float* h = H + (size_t)n * F;
    float s = 0.f, d = 0.f;
#pragma unroll 8
    for (int f = 0; f < F; ++f) {
        float v = h[f];
        s += v * a_src[f];
        d += v * a_dst[f];
    }
    As[n] = s;
    Ad[n] = d;
}

// ---------- init per-node state and agg = bias ----------
template <int F>
__global__ void init_node(unsigned* __restrict__ menc, float* __restrict__ z,
                          float* __restrict__ agg, const float* __restrict__ bias,
                          int N) {
    int i = blockIdx.x * blockDim.x + threadIdx.x;
    if (i < N) { menc[i] = 0u; z[i] = 0.f; }   // 0 < enc(-inf) for all reals
    if (i < N * F) agg[i] = bias[i & (F - 1)];
}

__device__ __forceinline__ void edge_sd(const int* __restrict__ ei, int i,
                                        int& s, int& d) {
    if (i < N_EDGES) { s = ei[i]; d = ei[N_EDGES + i]; }
    else             { s = d = i - N_EDGES; }            // self loops
}

// ---------- edge pass A: leaky-relu score, segment max ----------
__global__ void edge_score(const int* __restrict__ ei,
                           const float* __restrict__ As, const float* __restrict__ Ad,
                           float* __restrict__ ebuf, unsigned* __restrict__ menc) {
    int i = blockIdx.x * blockDim.x + threadIdx.x;
    if (i >= E_TOT) return;
    int s, d; edge_sd(ei, i, s, d);
    float e = As[s] + Ad[d];
    e = (e > 0.f) ? e : NEG_SLOPE * e;
    ebuf[i] = e;
    atomicMax(menc + d, enc_f32(e));
}

// ---------- edge pass B: exp(e - max), segment sum ----------
__global__ void edge_exp(const int* __restrict__ ei,
                         float* __restrict__ ebuf,
                         const unsigned* __restrict__ menc,
                         float* __restrict__ z) {
    int i = blockIdx.x * blockDim.x + threadIdx.x;
    if (i >= E_TOT) return;
    int s, d; edge_sd(ei, i, s, d);
    float ex = __expf(ebuf[i] - dec_f32(menc[d]));
    ebuf[i] = ex;
    atomicAdd(z + d, ex);
}

// ---------- edge pass C: agg[dst] += (ex/z[dst]) * h[src]  (wave per edge) ----------
template <int F>
__global__ void edge_agg(const int* __restrict__ ei,
                         const float* __restrict__ ebuf, const float* __restrict__ z,
                         const float* __restrict__ H, float* __restrict__ agg) {
    int g = blockIdx.x * blockDim.x + threadIdx.x;
    int edge = g >> 5;
    int lane = g & 31;
    if (edge >= E_TOT) return;
    int s, d; edge_sd(ei, edge, s, d);
    float coef = ebuf[edge] / z[d];
    const float* hs = H + (size_t)s * F;
    float* ad = agg + (size_t)d * F;
#pragma unroll
    for (int f = lane; f < F; f += 32)
        atomicAdd(ad + f, coef * hs[f]);
}

// ---------- elementwise relu ----------
__global__ void relu_kernel(float* __restrict__ A, int n) {
    int i = blockIdx.x * blockDim.x + threadIdx.x;
    if (i < n) A[i] = fmaxf(A[i], 0.f);
}

// ---------- log_softmax over rows of 32 (wave per node, lane = column) ----------
__global__ void log_softmax32(const float* __restrict__ A, float* __restrict__ out,
                              int N) {
    int g = blockIdx.x * blockDim.x + threadIdx.x;
    int n = g >> 5;
    int lane = g & 31;
    if (n >= N) return;
    float v = A[(size_t)n * 32 + lane];
    float m = v;
#pragma unroll
    for (int o = 16; o > 0; o >>= 1) m = fmaxf(m, __shfl_xor(m, o, 32));
    float ex = __expf(v - m);
    float sum = ex;
#pragma unroll
    for (int o = 16; o > 0; o >>= 1) sum += __shfl_xor(sum, o, 32);
    out[(size_t)n * 32 + lane] = v - m - __logf(sum);
}

static inline int cdiv(long long a, long long b) { return (int)((a + b - 1) / b); }

extern "C" void kernel_launch(void* const* d_in, const int* in_sizes, int n_in,
                              void* d_out, int out_size, void* d_ws, size_t ws_size,
                              hipStream_t stream) {
    const float* x      = (const float*)d_in[0];
    const int*   ei     = (const int*)  d_in[1];
    const float* W1     = (const float*)d_in[2];
    const float* a_s1   = (const float*)d_in[3];
    const float* a_d1   = (const float*)d_in[4];
    const float* b1     = (const float*)d_in[5];
    const float* W2     = (const float*)d_in[6];
    const float* a_s2   = (const float*)d_in[7];
    const float* a_d2   = (const float*)d_in[8];
    const float* b2     = (const float*)d_in[9];
    float* out          = (float*)d_out;

    // ---- workspace carving (floats) ----
    float*    ws   = (float*)d_ws;
    float*    h1   = ws;                                   // N * 64
    float*    agg1 = h1 + (size_t)N_NODES * HID_DIM;       // N * 64
    float*    As   = agg1 + (size_t)N_NODES * HID_DIM;     // N
    float*    Ad   = As + N_NODES;                         // N
    unsigned* menc = (unsigned*)(Ad + N_NODES);            // N
    float*    z    = (float*)menc + N_NODES;               // N
    float*    ebuf = z + N_NODES;                          // E_TOT
    float*    h2   = h1;    // safe alias: h1 dead after layer-1 finalize
    float*    agg2 = agg1;  // safe alias: agg1 dead after layer-2 GEMM reads it

    const int B = 256;

    // ================= Layer 1: x[N,128] -> agg1[N,64] =================
    {
        long long waves = (long long)(N_NODES / 16) * (HID_DIM / 16);
        gemm_wmma_f32<IN_DIM, HID_DIM>
            <<<cdiv(waves * 32, B), B, 0, stream>>>(x, W1, h1, N_NODES);
    }
    alpha_kernel<HID_DIM><<<cdiv(N_NODES, B), B, 0, stream>>>(h1, a_s1, a_d1, As, Ad, N_NODES);
    init_node<HID_DIM><<<cdiv((long long)N_NODES * HID_DIM, B), B, 0, stream>>>(menc, z, agg1, b1, N_NODES);
    edge_score<<<cdiv(E_TOT, B), B, 0, stream>>>(ei, As, Ad, ebuf, menc);
    edge_exp  <<<cdiv(E_TOT, B), B, 0, stream>>>(ei, ebuf, menc, z);
    edge_agg<HID_DIM><<<cdiv((long long)E_TOT * 32, B), B, 0, stream>>>(ei, ebuf, z, h1, agg1);
    relu_kernel<<<cdiv((long long)N_NODES * HID_DIM, B), B, 0, stream>>>(agg1, N_NODES * HID_DIM);

    // ================= Layer 2: agg1[N,64] -> agg2[N,32] =================
    {
        long long waves = (long long)(N_NODES / 16) * (OUT_DIM / 16);
        gemm_wmma_f32<HID_DIM, OUT_DIM>
            <<<cdiv(waves * 32, B), B, 0, stream>>>(agg1, W2, h2, N_NODES);
    }
    alpha_kernel<OUT_DIM><<<cdiv(N_NODES, B), B, 0, stream>>>(h2, a_s2, a_d2, As, Ad, N_NODES);
    init_node<OUT_DIM><<<cdiv((long long)N_NODES * OUT_DIM, B), B, 0, stream>>>(menc, z, agg2, b2, N_NODES);
    edge_score<<<cdiv(E_TOT, B), B, 0, stream>>>(ei, As, Ad, ebuf, menc);
    edge_exp  <<<cdiv(E_TOT, B), B, 0, stream>>>(ei, ebuf, menc, z);
    edge_agg<OUT_DIM><<<cdiv((long long)E_TOT * 32, B), B, 0, stream>>>(ei, ebuf, z, h2, agg2);

    // ================= log_softmax rows -> d_out =================
    log_softmax32<<<cdiv((long long)N_NODES * 32, B), B, 0, stream>>>(agg2, out, N_NODES);
}